// GCN_4346506904041
// MI455X (gfx1250) — compile-verified
//
#include <hip/hip_runtime.h>

// ---------------------------------------------------------------------------
// GCN inference on gfx1250 (MI455X), fp32 end-to-end using the CDNA5 native
// fp32 matrix op V_WMMA_F32_16X16X4_F32.
//
//   T  = X @ W1                        (WMMA, K=128)
//   H  = relu(A @ T + b1)              (WMMA, K=2048, split across 4 waves)
//   T  = H @ W2                        (WMMA, K=32)
//   H  = relu(A @ T + b2)              (WMMA, K=2048)
//   out= softmax(flat(H) @ Wf + bf)    (VALU + LDS reduce)
//
// Memory-bound on the 128MB adjacency (read twice; 2nd pass mostly L2-resident
// since A fits in the 192MB L2). fp32 WMMA compute (4.3 GFLOP) hides under it.
// ---------------------------------------------------------------------------

typedef float v2f __attribute__((ext_vector_type(2)));
typedef float v8f __attribute__((ext_vector_type(8)));

// 8-arg VOP3P form: (neg_a, A, neg_b, B, c_mod, C, reuse_a, reuse_b)
#define WMMA_F32(a, b, c) \
  __builtin_amdgcn_wmma_f32_16x16x4_f32(false, (a), false, (b), (short)0, (c), false, false)

// ---------------------------------------------------------------------------
// Fragment layout assumptions (ISA 7.12.2, 32-bit matrices, wave32):
//   A (16x4):  lane<16 : row = lane,    regs = {A[row][k+0], A[row][k+1]}
//              lane>=16: row = lane-16, regs = {A[row][k+2], A[row][k+3]}
//   B (4x16):  lane<16 : col = lane,    regs = {B[k+0][col], B[k+1][col]}
//              lane>=16: col = lane-16, regs = {B[k+2][col], B[k+3][col]}
//   C/D (16x16): reg v, lane<16 : (row=v,   col=lane)
//                reg v, lane>=16: (row=v+8, col=lane-16)
// ---------------------------------------------------------------------------

// T[M x 32] = X[M x ldX] @ W[K x 32].  One wave per 16-row tile.
__global__ __launch_bounds__(128) void gcn_transform(
    const float* __restrict__ X, const float* __restrict__ W,
    float* __restrict__ T, int Mtiles, int K, int ldX) {
  const int wave = blockIdx.x * (blockDim.x >> 5) + (threadIdx.x >> 5);
  if (wave >= Mtiles) return;  // wave-uniform: EXEC stays all-1s for WMMA
  const int lane = threadIdx.x & 31;
  const int half = lane >> 4, l16 = lane & 15;

  const float* aRow = X + (size_t)wave * 16 * ldX + (size_t)l16 * ldX + 2 * half;
  v8f acc0 = {}, acc1 = {};
  for (int k = 0; k < K; k += 4) {
    v2f a;
    a.x = aRow[k];
    a.y = aRow[k + 1];
    const float* bP = W + (size_t)(k + 2 * half) * 32;
    v2f b0, b1;
    b0.x = bP[l16];      b0.y = bP[32 + l16];       // cols 0..15
    b1.x = bP[16 + l16]; b1.y = bP[48 + l16];       // cols 16..31
    acc0 = WMMA_F32(a, b0, acc0);
    acc1 = WMMA_F32(a, b1, acc1);
  }
  float* out = T + (size_t)wave * 16 * 32;
  #pragma unroll
  for (int v = 0; v < 8; ++v) {
    out[(size_t)(v + 8 * half) * 32 + l16]      = acc0[v];
    out[(size_t)(v + 8 * half) * 32 + 16 + l16] = acc1[v];
  }
}

// Hout = relu(A @ Hin + bias). A: [B][2048][2048] f32, Hin/Hout: [B][2048][32].
// One block per 16-row output tile; 4 waves split K=2048 into 512-chunks,
// partials reduced through LDS.
__global__ __launch_bounds__(128) void gcn_aggregate(
    const float* __restrict__ A, const float* __restrict__ Hin,
    const float* __restrict__ bias, float* __restrict__ Hout) {
  __shared__ float part[4][16][33];  // +1 col pad vs LDS bank conflicts

  const int tile = blockIdx.x;       // 0 .. 8*128-1
  const int b = tile >> 7;           // batch
  const int r = tile & 127;          // 16-row tile within batch
  const int wid = threadIdx.x >> 5;  // wave id 0..3
  const int lane = threadIdx.x & 31;
  const int half = lane >> 4, l16 = lane & 15;
  const size_t N = 2048;

  const float* Ab = A + (size_t)b * N * N + (size_t)r * 16 * N;
  const float* Hb = Hin + (size_t)b * N * 32;
  const float* aRow = Ab + (size_t)l16 * N + 2 * half;
  const int k0 = wid * 512, k1 = k0 + 512;

  v8f acc0 = {}, acc1 = {};
  #pragma unroll 4
  for (int k = k0; k < k1; k += 4) {
    v2f a;
    a.x = aRow[k];                   // contiguous pair -> global_load_b64
    a.y = aRow[k + 1];
    const float* bP = Hb + (size_t)(k + 2 * half) * 32;
    v2f b0, b1;
    b0.x = bP[l16];      b0.y = bP[32 + l16];
    b1.x = bP[16 + l16]; b1.y = bP[48 + l16];
    acc0 = WMMA_F32(a, b0, acc0);
    acc1 = WMMA_F32(a, b1, acc1);
  }

  #pragma unroll
  for (int v = 0; v < 8; ++v) {
    part[wid][v + 8 * half][l16]      = acc0[v];
    part[wid][v + 8 * half][16 + l16] = acc1[v];
  }
  __syncthreads();

  float* out = Hout + (size_t)b * N * 32 + (size_t)r * 16 * 32;
  for (int e = threadIdx.x; e < 512; e += 128) {
    const int row = e >> 5, col = e & 31;
    float s = part[0][row][col] + part[1][row][col] +
              part[2][row][col] + part[3][row][col];
    s += bias[col];
    out[(size_t)row * 32 + col] = fmaxf(s, 0.0f);
  }
}

// out[b][0..9] = softmax(flat(H[b]) @ Wf + bf). One 256-thread block per batch.
__global__ __launch_bounds__(256) void gcn_head(
    const float* __restrict__ H, const float* __restrict__ Wf,
    const float* __restrict__ bf, float* __restrict__ out) {
  __shared__ float red[256][10];
  const int b = blockIdx.x, tid = threadIdx.x;
  const float* h = H + (size_t)b * 65536;

  float acc[10];
  #pragma unroll
  for (int c = 0; c < 10; ++c) acc[c] = 0.0f;

  for (int i = tid; i < 65536; i += 256) {
    const float x = h[i];
    const float* w = Wf + (size_t)i * 10;
    #pragma unroll
    for (int c = 0; c < 10; ++c) acc[c] = fmaf(x, w[c], acc[c]);
  }
  #pragma unroll
  for (int c = 0; c < 10; ++c) red[tid][c] = acc[c];
  __syncthreads();

  for (int s = 128; s > 0; s >>= 1) {
    if (tid < s) {
      #pragma unroll
      for (int c = 0; c < 10; ++c) red[tid][c] += red[tid + s][c];
    }
    __syncthreads();
  }

  if (tid == 0) {
    float logit[10], mx = -3.4e38f;
    #pragma unroll
    for (int c = 0; c < 10; ++c) {
      logit[c] = red[0][c] + bf[c];
      mx = fmaxf(mx, logit[c]);
    }
    float sum = 0.0f;
    #pragma unroll
    for (int c = 0; c < 10; ++c) {
      logit[c] = __expf(logit[c] - mx);
      sum += logit[c];
    }
    const float inv = 1.0f / sum;
    float* o = out + (size_t)b * 10;
    #pragma unroll
    for (int c = 0; c < 10; ++c) o[c] = logit[c] * inv;
  }
}

extern "C" void kernel_launch(void* const* d_in, const int* in_sizes, int n_in,
                              void* d_out, int out_size, void* d_ws, size_t ws_size,
                              hipStream_t stream) {
  (void)in_sizes; (void)n_in; (void)out_size; (void)ws_size;
  const float* adj = (const float*)d_in[0];  // [8,2048,2048]
  const float* X   = (const float*)d_in[1];  // [8,2048,128]
  const float* W1  = (const float*)d_in[2];  // [128,32]
  const float* b1  = (const float*)d_in[3];  // [32]
  const float* W2  = (const float*)d_in[4];  // [32,32]
  const float* b2  = (const float*)d_in[5];  // [32]
  const float* Wf  = (const float*)d_in[6];  // [65536,10]
  const float* bf  = (const float*)d_in[7];  // [10]
  float* out = (float*)d_out;                // [8,10]

  // Workspace: two 2MB fp32 slots (8*2048*32 each), ping-pong.
  float* T = (float*)d_ws;
  float* H = T + (size_t)8 * 2048 * 32;

  // 1) T = X @ W1          (16384 rows -> 1024 tiles, K=128)
  gcn_transform<<<256, 128, 0, stream>>>(X, W1, T, 1024, 128, 128);
  // 2) H = relu(A @ T + b1)
  gcn_aggregate<<<1024, 128, 0, stream>>>(adj, T, b1, H);
  // 3) T = H @ W2          (K=32)
  gcn_transform<<<256, 128, 0, stream>>>(H, W2, T, 1024, 32, 32);
  // 4) H = relu(A @ T + b2)
  gcn_aggregate<<<1024, 128, 0, stream>>>(adj, T, b2, H);
  // 5) out = softmax(flat(H) @ Wf + bf)
  gcn_head<<<8, 256, 0, stream>>>(H, Wf, bf, out);
}